// AI4DEM_33749853012359
// MI455X (gfx1250) — compile-verified
//
#include <hip/hip_runtime.h>
#include <hip/hip_bf16.h>
#include <stdint.h>

// ---------------------------------------------------------------------------
// DEM contact forces, 96^3 grid, one particle per occupied cell.
//   K1: zero a padded 100^3 AoS grid (8 floats/cell) in d_ws
//   K2: scatter particles (and wrap-around halo images) into the grid
//   K3: per 8^3 tile: stage the 12^3 tile+halo into LDS with the CDNA5
//       Tensor Data Mover (tensor_load_to_lds + s_wait_tensorcnt), then each
//       thread computes TWO x-adjacent cells' 125-neighbor force sums,
//       sharing overlapping neighbor loads (150 ds_load_b128 per cell).
// ---------------------------------------------------------------------------

#define GDIM 96      // logical grid cells per axis
#define HALO 2       // filter_size/2
#define PDIM 100     // padded grid dim
#define TILE 8       // cells per tile per axis
#define LDIM 12      // TILE + 2*HALO
#define NTILE 12     // GDIM / TILE
#define NTHREADS 256 // 4x8x8 cell-pairs per block

#define GLOBAL_AS __attribute__((address_space(1)))
#define LDS_AS    __attribute__((address_space(3)))

typedef int          v4i __attribute__((ext_vector_type(4)));
typedef unsigned int v4u __attribute__((ext_vector_type(4)));
typedef int          v8i __attribute__((ext_vector_type(8)));

#ifndef __has_builtin
#define __has_builtin(x) 0
#endif

#if __has_builtin(__builtin_amdgcn_tensor_load_to_lds)
#define HAVE_TDM 1
#else
#define HAVE_TDM 0
#endif

#if __has_builtin(__builtin_amdgcn_global_load_async_to_lds_b128)
#define HAVE_ASYNC_LDS 1
#else
#define HAVE_ASYNC_LDS 0
#endif

// Copy 16 bytes global -> LDS (per-lane async path, ASYNCcnt-tracked).
__device__ __forceinline__ void async_copy16(void* lds_dst, const void* gsrc) {
#if HAVE_ASYNC_LDS
    __builtin_amdgcn_global_load_async_to_lds_b128(
        (GLOBAL_AS v4i*)(void*)gsrc, (LDS_AS v4i*)lds_dst, 0, 0);
#else
    *(float4*)lds_dst = *(const float4*)gsrc;
#endif
}

__device__ __forceinline__ void wait_async_copies() {
#if HAVE_ASYNC_LDS
#if __has_builtin(__builtin_amdgcn_s_wait_asynccnt)
    __builtin_amdgcn_s_wait_asynccnt(0);
#else
    asm volatile("s_wait_asynccnt 0" ::: "memory");
#endif
#endif
}

// ---------------------------------------------------------------------------
// K3 (first in file so the disasm snippet shows it).
// Cell AoS layout: a = (x, y, z, vx), b = (vy, vz, idx+1, 0).
// ---------------------------------------------------------------------------
__global__ __launch_bounds__(NTHREADS)
void dem_forces(const float4* __restrict__ grid,
                const float* __restrict__ dp, const float* __restrict__ knp,
                const float* __restrict__ etap,
                float* __restrict__ out, int n) {
    __shared__ float4 tile[LDIM * LDIM * LDIM * 2];   // 55,296 bytes

    const int tid = threadIdx.x;
    const int ox0 = blockIdx.x * TILE;                // padded-space tile origin
    const int oy0 = blockIdx.y * TILE;
    const int oz0 = blockIdx.z * TILE;

#if HAVE_TDM
    // ---- Tensor Data Mover: one 3D tile DMA (12 x 12 rows of 96 dwords) ----
    if (tid < 32) {   // wave 0 issues; TDM ignores EXEC, so gate by wave branch
        const uint64_t gaddr = (uint64_t)(uintptr_t)grid
            + (uint64_t)(((oz0 * PDIM + oy0) * PDIM + ox0)) * 32u;
        const uint32_t laddr = (uint32_t)(size_t)(LDS_AS void*)&tile[0];
        // D# group 0: count=1 | lds_addr | global_addr(57b) | type=2
        v4u g0 = { 1u, laddr, (uint32_t)gaddr,
                   (uint32_t)((gaddr >> 32) & 0x01FFFFFFu) | (2u << 30) };
        // D# group 1: data_size=2 (4B); tensor_dim0=800, tensor_dim1=100;
        //             tile_dim0=96, tile_dim1=12, tile_dim2=12;
        //             dim0_stride=800, dim1_stride=80000 floats
        v8i g1 = { (int)(2u << 16),                          // data_size=4B
                   (int)(((PDIM * 8u) & 0xFFFFu) << 16),     // tensor_dim0 lo
                   (int)(((unsigned)PDIM) << 16),            // dim0 hi=0 | tensor_dim1 lo
                   (int)(((unsigned)(LDIM * 8u)) << 16),     // dim1 hi=0 | tile_dim0
                   (int)((unsigned)LDIM | ((unsigned)LDIM << 16)),  // tile_dim1|tile_dim2
                   (int)(PDIM * 8u),                         // tensor_dim0_stride lo32
                   (int)(((PDIM * PDIM * 8u) & 0xFFFFu) << 16), // d0s hi=0 | d1 stride lo16
                   (int)((PDIM * PDIM * 8u) >> 16) };        // dim1_stride bits 47:16
        // D# group 2: tensor_dim2=100, tensor_dim2_stride=8,000,000 floats
        v4i g2 = { PDIM, 0, (int)(PDIM * PDIM * PDIM * 8u), 0 };
        v4i g3 = { 0, 0, 0, 0 };
        v8i g4 = { 0, 0, 0, 0, 0, 0, 0, 0 };   // trailing group (6-arg form), unused
        __builtin_amdgcn_tensor_load_to_lds(g0, g1, g2, g3, g4, 0);
        __builtin_amdgcn_s_wait_tensorcnt(0);
    }
    __syncthreads();
#else
    // ---- fallback: per-lane async global->LDS copies ----
    const int total = LDIM * LDIM * LDIM * 2;   // 3456 float4's
    for (int t = tid; t < total; t += NTHREADS) {
        int cell = t >> 1, half = t & 1;
        int lx = cell % LDIM;
        int ly = (cell / LDIM) % LDIM;
        int lz = cell / (LDIM * LDIM);
        size_t gcell = ((size_t)(oz0 + lz) * PDIM + (size_t)(oy0 + ly)) * PDIM
                     + (size_t)(ox0 + lx);
        async_copy16(&tile[t], &grid[gcell * 2 + half]);
    }
    wait_async_copies();
    __syncthreads();
#endif

    // ---- per-thread: two x-adjacent central cells ----
    const float d    = dp[0];
    const float kn   = knp[0];
    const float eta  = etap[0];
    const float twod = 2.0f * d;

    const int tx = tid & 3;               // 0..3  -> cell pair along x
    const int ty = (tid >> 2) & 7;        // 0..7
    const int tz = tid >> 5;              // 0..7
    const int lx0 = HALO + (tx << 1);
    const int ly  = ty + HALO;
    const int lz  = tz + HALO;

    const int c0 = (lz * LDIM + ly) * LDIM + lx0;
    float4 a0 = tile[c0 * 2 + 0], b0 = tile[c0 * 2 + 1];
    float4 a1 = tile[c0 * 2 + 2], b1 = tile[c0 * 2 + 3];
    const int idx0 = (int)b0.z - 1;
    const int idx1 = (int)b1.z - 1;

    const float x0 = a0.x, y0 = a0.y, z0 = a0.z, vx0 = a0.w, vy0 = b0.x, vz0 = b0.y;
    const float x1 = a1.x, y1 = a1.y, z1 = a1.z, vx1 = a1.w, vy1 = b1.x, vz1 = b1.y;

    float fxc0 = 0.f, fyc0 = 0.f, fzc0 = 0.f, fxd0 = 0.f, fyd0 = 0.f, fzd0 = 0.f;
    float fxc1 = 0.f, fyc1 = 0.f, fzc1 = 0.f, fxd1 = 0.f, fyd1 = 0.f, fzd1 = 0.f;

    #pragma unroll 1
    for (int oz = -HALO; oz <= HALO; ++oz) {
        #pragma unroll 1
        for (int oy = -HALO; oy <= HALO; ++oy) {
            const int row = ((lz + oz) * LDIM + (ly + oy)) * LDIM;
            // 6 consecutive neighbor cells cover the x-stencils of both cells
            #pragma unroll
            for (int j = 0; j < 6; ++j) {
                const int nc = row + (lx0 - HALO + j);
                float4 na = tile[nc * 2 + 0];   // ds_load_b128
                float4 nb = tile[nc * 2 + 1];   // ds_load_b128
                if (j < 5) {  // neighbor of cell 0 (offsets -2..+2)
                    float dx = x0 - na.x, dy = y0 - na.y, dz = z0 - na.z;
                    float dvx = vx0 - na.w, dvy = vy0 - nb.x, dvz = vz0 - nb.y;
                    float dist = sqrtf(dx * dx + dy * dy + dz * dz);
                    float inv  = 1.0f / fmaxf(1e-4f, dist);
                    float vn   = (dvx * dx + dvy * dy + dvz * dz) * inv;
                    bool  ov   = dist < twod;
                    float fcol = ov ? kn  * (dist - twod) * inv : 0.f;
                    float fdmp = ov ? eta * vn * inv            : 0.f;
                    fxc0 += fcol * dx; fyc0 += fcol * dy; fzc0 += fcol * dz;
                    fxd0 += fdmp * dx; fyd0 += fdmp * dy; fzd0 += fdmp * dz;
                }
                if (j > 0) {  // neighbor of cell 1 (offsets -2..+2)
                    float dx = x1 - na.x, dy = y1 - na.y, dz = z1 - na.z;
                    float dvx = vx1 - na.w, dvy = vy1 - nb.x, dvz = vz1 - nb.y;
                    float dist = sqrtf(dx * dx + dy * dy + dz * dz);
                    float inv  = 1.0f / fmaxf(1e-4f, dist);
                    float vn   = (dvx * dx + dvy * dy + dvz * dz) * inv;
                    bool  ov   = dist < twod;
                    float fcol = ov ? kn  * (dist - twod) * inv : 0.f;
                    float fdmp = ov ? eta * vn * inv            : 0.f;
                    fxc1 += fcol * dx; fyc1 += fcol * dy; fzc1 += fcol * dz;
                    fxd1 += fdmp * dx; fyd1 += fdmp * dy; fzd1 += fdmp * dz;
                }
            }
        }
    }

    // wall forces, domain extent = GDIM * d
    const float dsx = (float)GDIM * d;
    auto wall = [&](float p) -> float {
        float lo = (p != 0.0f && p < d)  ? kn * (d - p)          : 0.f;
        float hi = (p > dsx - twod)      ? kn * (p - dsx + twod) : 0.f;
        return lo - hi;
    };

    const size_t N = (size_t)n;
    if (idx0 >= 0) {
        out[0 * N + idx0] = fxc0; out[1 * N + idx0] = fyc0; out[2 * N + idx0] = fzc0;
        out[3 * N + idx0] = fxd0; out[4 * N + idx0] = fyd0; out[5 * N + idx0] = fzd0;
        out[6 * N + idx0] = wall(x0); out[7 * N + idx0] = wall(y0); out[8 * N + idx0] = wall(z0);
    }
    if (idx1 >= 0) {
        out[0 * N + idx1] = fxc1; out[1 * N + idx1] = fyc1; out[2 * N + idx1] = fzc1;
        out[3 * N + idx1] = fxd1; out[4 * N + idx1] = fyd1; out[5 * N + idx1] = fzd1;
        out[6 * N + idx1] = wall(x1); out[7 * N + idx1] = wall(y1); out[8 * N + idx1] = wall(z1);
    }
}

// ---------------------------------------------------------------------------
// K1: zero the padded AoS grid (2 float4 per cell).
// ---------------------------------------------------------------------------
__global__ void dem_init_grid(float4* __restrict__ grid, int nvec) {
    int i = blockIdx.x * blockDim.x + threadIdx.x;
    if (i < nvec) grid[i] = make_float4(0.f, 0.f, 0.f, 0.f);
}

// ---------------------------------------------------------------------------
// K2: scatter particles into the padded grid (incl. wrap halo images).
// ---------------------------------------------------------------------------
__global__ void dem_scatter(const float* __restrict__ x, const float* __restrict__ y,
                            const float* __restrict__ z, const float* __restrict__ vx,
                            const float* __restrict__ vy, const float* __restrict__ vz,
                            const float* __restrict__ dp,
                            float4* __restrict__ grid, int n) {
    int i = blockIdx.x * blockDim.x + threadIdx.x;
    if (i >= n) return;
    float d  = dp[0];
    float xi = x[i], yi = y[i], zi = z[i];
    float invd = 1.0f / d;
    int cx = (int)rintf(xi * invd);
    int cy = (int)rintf(yi * invd);
    int cz = (int)rintf(zi * invd);
    float4 a = make_float4(xi, yi, zi, vx[i]);
    float4 b = make_float4(vy[i], vz[i], (float)(i + 1), 0.f);
    #pragma unroll
    for (int sz = -1; sz <= 1; ++sz) {
        int pz = cz + HALO + sz * GDIM;
        if (pz < 0 || pz >= PDIM) continue;
        #pragma unroll
        for (int sy = -1; sy <= 1; ++sy) {
            int py = cy + HALO + sy * GDIM;
            if (py < 0 || py >= PDIM) continue;
            #pragma unroll
            for (int sx = -1; sx <= 1; ++sx) {
                int px = cx + HALO + sx * GDIM;
                if (px < 0 || px >= PDIM) continue;
                size_t cell = ((size_t)pz * PDIM + (size_t)py) * PDIM + (size_t)px;
                grid[cell * 2 + 0] = a;
                grid[cell * 2 + 1] = b;
            }
        }
    }
}

// ---------------------------------------------------------------------------
// Host-side launcher (graph-capture safe: kernels only, all on `stream`).
// ---------------------------------------------------------------------------
extern "C" void kernel_launch(void* const* d_in, const int* in_sizes, int n_in,
                              void* d_out, int out_size, void* d_ws, size_t ws_size,
                              hipStream_t stream) {
    const float* x   = (const float*)d_in[0];
    const float* y   = (const float*)d_in[1];
    const float* z   = (const float*)d_in[2];
    const float* vx  = (const float*)d_in[3];
    const float* vy  = (const float*)d_in[4];
    const float* vz  = (const float*)d_in[5];
    const float* dp  = (const float*)d_in[6];   // d
    const float* knp = (const float*)d_in[7];   // kn
    const float* etp = (const float*)d_in[8];   // damping_coefficient_Eta
    // d_in[9]=friction (unused), d_in[10]=dt (unused),
    // d_in[11]=input_shape (hardcoded 96^3), d_in[12]=filter_size (5)

    const int n = in_sizes[0];
    float4* grid = (float4*)d_ws;               // needs PDIM^3 * 32 B = 32 MB
    (void)ws_size; (void)n_in; (void)out_size;

    const int nvec = PDIM * PDIM * PDIM * 2;    // 2,000,000 float4's
    dem_init_grid<<<(nvec + 255) / 256, 256, 0, stream>>>(grid, nvec);
    dem_scatter<<<(n + 255) / 256, 256, 0, stream>>>(x, y, z, vx, vy, vz, dp, grid, n);
    dim3 fgrid(NTILE, NTILE, NTILE);
    dem_forces<<<fgrid, NTHREADS, 0, stream>>>(grid, dp, knp, etp, (float*)d_out, n);
}